// GCNNet_38783554683010
// MI455X (gfx1250) — compile-verified
//
#include <hip/hip_runtime.h>
#include <hip/hip_bf16.h>
#include <math.h>

#define N_NODES 50000
#define N_EDGES 600000
#define N_GRAPHS 500
#define IN_DIM 64
#define HID 128
#define N_CLASSES 10
#define N_PROTO 50
#define PROTO_NODES 10
#define EPS_D 1e-4f

typedef float v2f __attribute__((ext_vector_type(2)));
typedef float v8f __attribute__((ext_vector_type(8)));

// ---------------------------------------------------------------------------
// Init: deg = 1.0 (self loop), gsum = 0, counts = 0
// ---------------------------------------------------------------------------
__global__ __launch_bounds__(256) void k_init(float* deg, float* gsum, float* counts) {
    int i = blockIdx.x * blockDim.x + threadIdx.x;
    if (i < N_NODES) deg[i] = 1.0f;
    if (i < N_GRAPHS * HID) gsum[i] = 0.0f;
    if (i < N_GRAPHS) counts[i] = 0.0f;
}

// deg[dst] += 1 per edge
__global__ __launch_bounds__(256) void k_degree(const int* __restrict__ edges, float* deg) {
    int e = blockIdx.x * blockDim.x + threadIdx.x;
    if (e >= N_EDGES) return;
    int dst = edges[N_EDGES + e];
    unsafeAtomicAdd(&deg[dst], 1.0f);
}

// dis = rsqrt(deg) in place
__global__ __launch_bounds__(256) void k_dis(float* deg) {
    int i = blockIdx.x * blockDim.x + threadIdx.x;
    if (i < N_NODES) deg[i] = rsqrtf(deg[i]);
}

// ---------------------------------------------------------------------------
// fp32 WMMA GEMM:  HW[N, 128] = A[N, K] @ W[K, 128]
// One wave per 16x16 output tile; 8 waves/block cover the 128 output columns
// of one 16-row stripe. K-loop of v_wmma_f32_16x16x4_f32.
//
// A-matrix VGPR layout (ISA 7.12.2, 32-bit A 16x4): lanes 0-15 hold M=lane,
// K={k,k+1}; lanes 16-31 hold M=lane-16, K={k+2,k+3}.
// B-matrix (4x16): row striped across lanes within a VGPR (mirror of A):
// lanes 0-15: N=lane, K={k,k+1}; lanes 16-31: N=lane-16, K={k+2,k+3}.
// C/D: VGPR v, lanes 0-15 -> (M=v, N=lane); lanes 16-31 -> (M=v+8, N=lane-16).
// ---------------------------------------------------------------------------
template <int K>
__global__ __launch_bounds__(256) void k_gemm_wmma(const float* __restrict__ A,
                                                   const float* __restrict__ W,
                                                   float* __restrict__ HW) {
    const int wave  = threadIdx.x >> 5;   // 0..7 -> output col tile
    const int lane  = threadIdx.x & 31;
    const int lhalf = lane & 15;
    const int hi    = lane >> 4;          // 0 or 1
    const int row0  = blockIdx.x * 16;    // 50000 % 16 == 0, no bounds check
    const int n0    = wave * 16;

    v8f acc = {};
#pragma unroll
    for (int k = 0; k < K; k += 4) {
        // A fragment
        const float* ap = A + (size_t)(row0 + lhalf) * K + (k + hi * 2);
        v2f a;
        a.x = ap[0];
        a.y = ap[1];
        // B fragment
        const float* wp = W + (size_t)(k + hi * 2) * HID + (n0 + lhalf);
        v2f b;
        b.x = wp[0];
        b.y = wp[HID];
        acc = __builtin_amdgcn_wmma_f32_16x16x4_f32(false, a, false, b,
                                                    (short)0, acc, false, false);
    }
#pragma unroll
    for (int v = 0; v < 8; ++v) {
        HW[(size_t)(row0 + hi * 8 + v) * HID + (n0 + lhalf)] = acc[v];
    }
}

// agg init with self-loop: AGG[i,c] = HW[i,c] * dis[i]^2
__global__ __launch_bounds__(256) void k_selfinit(const float* __restrict__ hw,
                                                  const float* __restrict__ dis,
                                                  float* __restrict__ agg) {
    int i = blockIdx.x * blockDim.x + threadIdx.x;
    if (i >= N_NODES * HID) return;
    float d = dis[i >> 7];  // i / 128
    agg[i] = hw[i] * d * d;
}

// edge scatter: AGG[dst] += HW[src] * dis[src]*dis[dst], float4 per thread
__global__ __launch_bounds__(256) void k_scatter(const int* __restrict__ edges,
                                                 const float* __restrict__ dis,
                                                 const float* __restrict__ hw,
                                                 float* __restrict__ agg) {
    int t = blockIdx.x * blockDim.x + threadIdx.x;
    if (t >= N_EDGES * 32) return;
    int e  = t >> 5;
    int c4 = (t & 31) << 2;
    int src = edges[e];
    int dst = edges[N_EDGES + e];
    float norm = dis[src] * dis[dst];
    const float4 v = *(const float4*)(hw + (size_t)src * HID + c4);
    float* out = agg + (size_t)dst * HID + c4;
    unsafeAtomicAdd(out + 0, v.x * norm);
    unsafeAtomicAdd(out + 1, v.y * norm);
    unsafeAtomicAdd(out + 2, v.z * norm);
    unsafeAtomicAdd(out + 3, v.w * norm);
}

// h = relu(agg + b), in place
__global__ __launch_bounds__(256) void k_bias_relu(float* __restrict__ h,
                                                   const float* __restrict__ b) {
    int i = blockIdx.x * blockDim.x + threadIdx.x;
    if (i >= N_NODES * HID) return;
    h[i] = fmaxf(h[i] + b[i & (HID - 1)], 0.0f);
}

// mean-readout accumulation: gsum[batch[n]] += h[n], counts[batch[n]] += 1
__global__ __launch_bounds__(256) void k_readout(const int* __restrict__ batch,
                                                 const float* __restrict__ h,
                                                 float* __restrict__ gsum,
                                                 float* __restrict__ counts) {
    int t = blockIdx.x * blockDim.x + threadIdx.x;
    if (t >= N_NODES * 32) return;
    int node = t >> 5;
    int c4   = (t & 31) << 2;
    int g = batch[node];
    const float4 v = *(const float4*)(h + (size_t)node * HID + c4);
    float* out = gsum + (size_t)g * HID + c4;
    unsafeAtomicAdd(out + 0, v.x);
    unsafeAtomicAdd(out + 1, v.y);
    unsafeAtomicAdd(out + 2, v.z);
    unsafeAtomicAdd(out + 3, v.w);
    if ((t & 31) == 0) unsafeAtomicAdd(&counts[g], 1.0f);
}

// graph_emb = gsum / max(counts, 1), in place
__global__ __launch_bounds__(256) void k_graphmean(float* __restrict__ gsum,
                                                   const float* __restrict__ counts) {
    int i = blockIdx.x * blockDim.x + threadIdx.x;
    if (i >= N_GRAPHS * HID) return;
    gsum[i] /= fmaxf(counts[i >> 7], 1.0f);
}

// proto_graph[p,c] = mean_j proto_node_emb[p,j,c]
__global__ __launch_bounds__(256) void k_proto(const float* __restrict__ pne,
                                               float* __restrict__ pg) {
    int t = blockIdx.x * blockDim.x + threadIdx.x;
    if (t >= N_PROTO * HID) return;
    int p = t >> 7;
    int c = t & (HID - 1);
    float s = 0.0f;
#pragma unroll
    for (int j = 0; j < PROTO_NODES; ++j)
        s += pne[((size_t)p * PROTO_NODES + j) * HID + c];
    pg[t] = s * (1.0f / PROTO_NODES);
}

// per-graph: d2 vs all prototypes, sim = log((d2+1)/(d2+eps)),
// logits[c] = sum_p sim[p] * last_w[c,p]
__global__ __launch_bounds__(64) void k_final(const float* __restrict__ gemb,
                                              const float* __restrict__ pg,
                                              const float* __restrict__ last_w,
                                              float* __restrict__ logits) {
    __shared__ float g[HID];
    __shared__ float sim[N_PROTO];
    int b = blockIdx.x;
    int tid = threadIdx.x;
    g[tid]      = gemb[(size_t)b * HID + tid];
    g[tid + 64] = gemb[(size_t)b * HID + tid + 64];
    __syncthreads();
    if (tid < N_PROTO) {
        float d2 = 0.0f;
        const float* pp = pg + (size_t)tid * HID;
        for (int c = 0; c < HID; ++c) {
            float diff = g[c] - pp[c];
            d2 = fmaf(diff, diff, d2);
        }
        d2 = fmaxf(d2, 0.0f);
        sim[tid] = logf((d2 + 1.0f) / (d2 + EPS_D));
    }
    __syncthreads();
    if (tid < N_CLASSES) {
        float acc = 0.0f;
        for (int p = 0; p < N_PROTO; ++p)
            acc += sim[p] * last_w[tid * N_PROTO + p];
        logits[(size_t)b * N_CLASSES + tid] = acc;
    }
}

// ---------------------------------------------------------------------------
static inline int cdiv(long long a, int b) { return (int)((a + b - 1) / b); }

extern "C" void kernel_launch(void* const* d_in, const int* in_sizes, int n_in,
                              void* d_out, int out_size, void* d_ws, size_t ws_size,
                              hipStream_t stream) {
    const float* x      = (const float*)d_in[0];
    const int*   edges  = (const int*)d_in[1];   // [2, E]: row0 src, row1 dst
    const int*   batch  = (const int*)d_in[2];
    const float* W0     = (const float*)d_in[3];
    const float* b0     = (const float*)d_in[4];
    const float* W1     = (const float*)d_in[5];
    const float* b1     = (const float*)d_in[6];
    const float* W2     = (const float*)d_in[7];
    const float* b2     = (const float*)d_in[8];
    const float* pne    = (const float*)d_in[9];
    const float* last_w = (const float*)d_in[10];
    float* logits = (float*)d_out;

    // workspace carve-up (floats)
    float* ws    = (float*)d_ws;
    float* dis   = ws;                        // 50000 (deg, then rsqrt in place)
    float* buf0  = dis + 50048;               // 50000*128  (HW)
    float* buf1  = buf0 + (size_t)N_NODES * HID;  // 50000*128  (AGG / h)
    float* gsum  = buf1 + (size_t)N_NODES * HID;  // 500*128
    float* cnt   = gsum + N_GRAPHS * HID;     // 500 (pad 512)
    float* pg    = cnt + 512;                 // 50*128

    const int NE  = N_NODES * HID;            // 6.4M elementwise
    const int ESC = N_EDGES * 32;             // scatter threads
    const int RSC = N_NODES * 32;             // readout threads
    const int MT  = N_NODES / 16;             // 3125 GEMM row tiles

    k_init<<<cdiv(N_GRAPHS * HID, 256), 256, 0, stream>>>(dis, gsum, cnt);
    k_degree<<<cdiv(N_EDGES, 256), 256, 0, stream>>>(edges, dis);
    k_dis<<<cdiv(N_NODES, 256), 256, 0, stream>>>(dis);

    // ---- layer 0: in = x [N,64] ----
    k_gemm_wmma<IN_DIM><<<MT, 256, 0, stream>>>(x, W0, buf0);
    k_selfinit<<<cdiv(NE, 256), 256, 0, stream>>>(buf0, dis, buf1);
    k_scatter<<<cdiv(ESC, 256), 256, 0, stream>>>(edges, dis, buf0, buf1);
    k_bias_relu<<<cdiv(NE, 256), 256, 0, stream>>>(buf1, b0);

    // ---- layer 1: in = buf1 [N,128] ----
    k_gemm_wmma<HID><<<MT, 256, 0, stream>>>(buf1, W1, buf0);
    k_selfinit<<<cdiv(NE, 256), 256, 0, stream>>>(buf0, dis, buf1);
    k_scatter<<<cdiv(ESC, 256), 256, 0, stream>>>(edges, dis, buf0, buf1);
    k_bias_relu<<<cdiv(NE, 256), 256, 0, stream>>>(buf1, b1);

    // ---- layer 2 ----
    k_gemm_wmma<HID><<<MT, 256, 0, stream>>>(buf1, W2, buf0);
    k_selfinit<<<cdiv(NE, 256), 256, 0, stream>>>(buf0, dis, buf1);
    k_scatter<<<cdiv(ESC, 256), 256, 0, stream>>>(edges, dis, buf0, buf1);
    k_bias_relu<<<cdiv(NE, 256), 256, 0, stream>>>(buf1, b2);

    // ---- readout + head ----
    k_readout<<<cdiv(RSC, 256), 256, 0, stream>>>(batch, buf1, gsum, cnt);
    k_graphmean<<<cdiv(N_GRAPHS * HID, 256), 256, 0, stream>>>(gsum, cnt);
    k_proto<<<cdiv(N_PROTO * HID, 256), 256, 0, stream>>>(pne, pg);
    k_final<<<N_GRAPHS, 64, 0, stream>>>(gsum, pg, last_w, logits);

    (void)in_sizes; (void)n_in; (void)out_size; (void)ws_size;
}